// SampledSoftmaxLoss_21655225106694
// MI455X (gfx1250) — compile-verified
//
#include <hip/hip_runtime.h>
#include <hip/hip_bf16.h>

typedef __attribute__((ext_vector_type(2))) float v2f;
typedef __attribute__((ext_vector_type(4))) float v4f;
typedef __attribute__((ext_vector_type(8))) float v8f;

#define D_MODEL 128
#define NUM_NEG 128

// One wave (32 lanes) per batch row.
// neg_sim via V_WMMA_F32_16X16X4_F32 (exact f32 math), broadcast-B trick:
//   A[m,k] = gathered negative row m (16 rows = lane%16), B[k,n] = user row
//   chunk broadcast over N, so C[m,n] = dot-chunk(neg[m], user) for all n.
// K-slot remapping for wide loads: each lane loads a 16B v4f covering dims
// [8c + 4*half .. +3]; WMMA #0 of the pair consumes .xy, WMMA #1 consumes .zw.
// A dot product is K-order invariant, and A/B use the same dim->slot map, so
// the two WMMAs together cover dims 8c..8c+7 exactly once. This halves vmem.
// Result: lanes 0-15 hold negs {t*16+0..7} in acc[t][0..7], lanes 16-31 hold
// negs {t*16+8..15}; the two halves' logsumexp partials merge via shfl_xor 16.
__global__ __launch_bounds__(256)
void sampled_softmax_rows(const float* __restrict__ user,       // (B, 128)
                          const float* __restrict__ items,      // (C, 128)
                          const long long* __restrict__ pos_ids,// (B,)
                          const long long* __restrict__ neg_ids,// (B, 128)
                          float* __restrict__ row_loss,         // (B,)
                          int batch)
{
    const int lane = threadIdx.x & 31;
    const int wave = threadIdx.x >> 5;
    const int row  = blockIdx.x * 8 + wave;
    if (row >= batch) return;            // wave-uniform: EXEC all-ones for WMMA

    const int half = lane >> 4;          // 0: lanes 0-15 / 1: lanes 16-31
    const int m    = lane & 15;          // A-matrix M slot

    // Gathered base pointers for this lane's negative rows (8 N-tiles).
    const float* nrow[8];
#pragma unroll
    for (int t = 0; t < 8; ++t) {
        long long idx = neg_ids[(long long)row * NUM_NEG + t * 16 + m];
        nrow[t] = items + idx * (long long)D_MODEL;
    }
    const float* urow = user + (long long)row * D_MODEL;
    const float* prow = items + pos_ids[row] * (long long)D_MODEL;

    v8f zero = {};
    v8f acc[8];
#pragma unroll
    for (int t = 0; t < 8; ++t) acc[t] = zero;

    float psum = 0.0f;

#pragma unroll 2
    for (int c = 0; c < D_MODEL / 8; ++c) {          // 16 chunks of 8 dims
        const int d0 = c * 8 + half * 4;             // 16B-aligned per lane half
        v4f uu = *(const v4f*)(urow + d0);           // B operand (broadcast over N)
        v4f pp = *(const v4f*)(prow + d0);
        psum += uu.x * pp.x + uu.y * pp.y + uu.z * pp.z + uu.w * pp.w;

        v2f u_lo = __builtin_shufflevector(uu, uu, 0, 1);
        v2f u_hi = __builtin_shufflevector(uu, uu, 2, 3);

        v4f aa[8];
#pragma unroll
        for (int t = 0; t < 8; ++t)
            aa[t] = *(const v4f*)(nrow[t] + d0);     // gathered A operand, b128

#pragma unroll
        for (int t = 0; t < 8; ++t) {                // 8 independent WMMAs
            v2f a_lo = __builtin_shufflevector(aa[t], aa[t], 0, 1);
            acc[t] = __builtin_amdgcn_wmma_f32_16x16x4_f32(
                false, a_lo, false, u_lo, (short)0, acc[t], false, false);
        }
#pragma unroll
        for (int t = 0; t < 8; ++t) {                // second K pair, same regs
            v2f a_hi = __builtin_shufflevector(aa[t], aa[t], 2, 3);
            acc[t] = __builtin_amdgcn_wmma_f32_16x16x4_f32(
                false, a_hi, false, u_hi, (short)0, acc[t], false, false);
        }
    }

    // pos_sim: lane half 0 covers dims 8c..8c+3, half 1 covers 8c+4..8c+7
    psum += __shfl_xor(psum, 16, 32);
    const float pos = psum;   // TEMPERATURE == 1

    // logsumexp over this lane-half's 64 neg logits (the two halves run in
    // parallel lanes -> 64 exp instructions cover all 128 logits)
    float mx = -INFINITY;
#pragma unroll
    for (int t = 0; t < 8; ++t)
#pragma unroll
        for (int r = 0; r < 8; ++r)
            mx = fmaxf(mx, acc[t][r]);
    float se = 0.0f;
#pragma unroll
    for (int t = 0; t < 8; ++t)
#pragma unroll
        for (int r = 0; r < 8; ++r)
            se += __expf(acc[t][r] - mx);

    // merge the two lane-halves + the positive logit
    const float mo = __shfl_xor(mx, 16, 32);
    const float so = __shfl_xor(se, 16, 32);
    const float M  = fmaxf(fmaxf(mx, mo), pos);
    const float st = se * __expf(mx - M) + so * __expf(mo - M) + __expf(pos - M);
    const float loss = __logf(st) + M - pos;

    if (lane == 0) row_loss[row] = loss;
}

// Deterministic single-block mean reduction (no float atomics).
__global__ __launch_bounds__(256)
void reduce_mean(const float* __restrict__ row_loss, float* __restrict__ out, int batch)
{
    __shared__ float sm[256];
    float s = 0.0f;
    for (int i = threadIdx.x; i < batch; i += 256) s += row_loss[i];
    sm[threadIdx.x] = s;
    __syncthreads();
    for (int w = 128; w > 0; w >>= 1) {
        if ((int)threadIdx.x < w) sm[threadIdx.x] += sm[threadIdx.x + w];
        __syncthreads();
    }
    if (threadIdx.x == 0) out[0] = sm[0] / (float)batch;
}

extern "C" void kernel_launch(void* const* d_in, const int* in_sizes, int n_in,
                              void* d_out, int out_size, void* d_ws, size_t ws_size,
                              hipStream_t stream)
{
    const float*     user  = (const float*)d_in[0];      // (B, 128) f32
    const float*     items = (const float*)d_in[1];      // (C, 128) f32
    const long long* pos   = (const long long*)d_in[2];  // (B,) i64
    const long long* neg   = (const long long*)d_in[3];  // (B, 128) i64
    const int batch = in_sizes[2];

    float* row_loss = (float*)d_ws;                      // batch floats of scratch

    const int blocks = (batch + 7) / 8;                  // 8 waves (rows) per block
    sampled_softmax_rows<<<blocks, 256, 0, stream>>>(user, items, pos, neg, row_loss, batch);
    reduce_mean<<<1, 256, 0, stream>>>(row_loss, (float*)d_out, batch);
}